// SSLPretrainModel_72310069395907
// MI455X (gfx1250) — compile-verified
//
#include <hip/hip_runtime.h>
#include <hip/hip_bf16.h>

#define N_ATOMS 200000
#define E_EDGES 800000
#define HDIM    512
#define AF      133
#define BF      14
#define B_GRAPHS 4096

typedef __attribute__((ext_vector_type(2))) float v2f;
typedef __attribute__((ext_vector_type(8))) float v8f;

__device__ __forceinline__ v8f wmma_f32_4(v2f a, v2f b, v8f c) {
    // 8-arg form: (neg_a, A, neg_b, B, c_mod, C, reuse_a, reuse_b)
    return __builtin_amdgcn_wmma_f32_16x16x4_f32(false, a, false, b,
                                                 (short)0, c, false, false);
}

__device__ __forceinline__ v8f v8zero() {
    v8f z;
#pragma unroll
    for (int i = 0; i < 8; ++i) z[i] = 0.0f;
    return z;
}

// ---------------------------------------------------------------------------
// node_pred = A @ W_node + b_node     [N_ATOMS x AF]
// 8 waves/block, each wave = 16 rows, block = 128 rows. K blocked by 64,
// W tile staged in LDS (64 x 144 floats, cols padded 133->144).
// ---------------------------------------------------------------------------
__global__ __launch_bounds__(256) void node_kernel(
        const float* __restrict__ A, const float* __restrict__ Wn,
        const float* __restrict__ bn, float* __restrict__ out_node) {
    __shared__ float ldsW[64 * 144];
    const int tid  = threadIdx.x;
    const int wave = tid >> 5;
    const int lane = tid & 31;
    const int m    = lane & 15;          // row-in-tile (A) / col-in-tile (B,D)
    const int koff = (lane >> 4) << 1;   // K sub-offset: 0 or 2
    const int row16 = blockIdx.x * 128 + wave * 16;

    v8f acc[9];
#pragma unroll
    for (int t = 0; t < 9; ++t) acc[t] = v8zero();

    // zero once: padded cols 133..143 stay zero for all K blocks
    for (int i = tid; i < 64 * 144; i += 256) ldsW[i] = 0.0f;

    for (int kb = 0; kb < 8; ++kb) {
        __syncthreads();
        for (int i = tid; i < 64 * AF; i += 256) {
            int r = i / AF, c = i - r * AF;
            ldsW[r * 144 + c] = Wn[(kb * 64 + r) * AF + c];
        }
        __syncthreads();

        // A fragments for this K block (16 k-steps of 4)
        v2f afr[16];
        const int row = row16 + m;
#pragma unroll
        for (int ks = 0; ks < 16; ++ks) {
            int kg = kb * 64 + ks * 4 + koff;
            if (row < N_ATOMS) {
                afr[ks] = *reinterpret_cast<const v2f*>(A + (long)row * HDIM + kg);
            } else {
                afr[ks][0] = 0.0f; afr[ks][1] = 0.0f;
            }
        }

        for (int nt = 0; nt < 9; ++nt) {
            const int col = nt * 16 + m;
#pragma unroll
            for (int ks = 0; ks < 16; ++ks) {
                int kl = ks * 4 + koff;
                v2f bfr;
                bfr[0] = ldsW[kl * 144 + col];
                bfr[1] = ldsW[(kl + 1) * 144 + col];
                acc[nt] = wmma_f32_4(afr[ks], bfr, acc[nt]);
            }
        }
    }

    // D layout: VGPR i -> M = i (+8 for lanes 16..31), N = lane&15
    const int rbump = (lane >> 4) << 3;
#pragma unroll
    for (int nt = 0; nt < 9; ++nt) {
        int col = nt * 16 + m;
        if (col < AF) {
            float bias = bn[col];
#pragma unroll
            for (int i = 0; i < 8; ++i) {
                int row = row16 + i + rbump;
                if (row < N_ATOMS)
                    out_node[(long)row * AF + col] = acc[nt][i] + bias;
            }
        }
    }
}

// ---------------------------------------------------------------------------
// edge_pred = 0.5*(A[a1]+A[a2]) @ W_edge + b_edge   [E/2 x BF]
// kept edge k corresponds to directed edge 2k (rev = e^1 => keep even e).
// W_edge fully staged in LDS (512 x 16, cols padded 14->16).
// ---------------------------------------------------------------------------
__global__ __launch_bounds__(256) void edge_kernel(
        const float* __restrict__ A, const int* __restrict__ eidx,
        const float* __restrict__ We, const float* __restrict__ be,
        float* __restrict__ out_edge) {
    __shared__ float ldsW[HDIM * 16];
    const int tid  = threadIdx.x;
    const int wave = tid >> 5;
    const int lane = tid & 31;
    const int m    = lane & 15;
    const int koff = (lane >> 4) << 1;

    for (int i = tid; i < HDIM * 16; i += 256) ldsW[i] = 0.0f;
    __syncthreads();
    for (int i = tid; i < HDIM * BF; i += 256) {
        int r = i / BF, c = i - r * BF;
        ldsW[r * 16 + c] = We[r * BF + c];
    }
    __syncthreads();

    const long kbase = (long)blockIdx.x * 128 + wave * 16;   // kept-edge tile base
    const long e = kbase + m;                                 // grid sized exactly
    const int a1 = eidx[2 * e];              // edge_index[0, 2k]
    const int a2 = eidx[E_EDGES + 2 * e];    // edge_index[1, 2k]
    const float* r1 = A + (long)a1 * HDIM;
    const float* r2 = A + (long)a2 * HDIM;

    v8f acc = v8zero();
#pragma unroll 4
    for (int ks = 0; ks < 128; ++ks) {
        int kl = ks * 4 + koff;
        v2f x1 = *reinterpret_cast<const v2f*>(r1 + kl);
        v2f x2 = *reinterpret_cast<const v2f*>(r2 + kl);
        v2f a; a[0] = 0.5f * (x1[0] + x2[0]); a[1] = 0.5f * (x1[1] + x2[1]);
        v2f b; b[0] = ldsW[kl * 16 + m]; b[1] = ldsW[(kl + 1) * 16 + m];
        acc = wmma_f32_4(a, b, acc);
    }

    if (m < BF) {
        float bias = be[m];
        const int rbump = (lane >> 4) << 3;
#pragma unroll
        for (int i = 0; i < 8; ++i) {
            long k = kbase + i + rbump;
            out_edge[k * BF + m] = acc[i] + bias;
        }
    }
}

// ---------------------------------------------------------------------------
// graph_emb[g] = sum of A rows whose (sorted) batch id == g   [B x H] into ws
// Deterministic (fixed-order) segmented reduction, one block per graph.
// ---------------------------------------------------------------------------
__global__ __launch_bounds__(256) void segsum_kernel(
        const float* __restrict__ A, const int* __restrict__ batch,
        float* __restrict__ gemb) {
    const int g = blockIdx.x;
    int lo = 0, hi = N_ATOMS;
    while (lo < hi) { int mid = (lo + hi) >> 1; if (batch[mid] < g) lo = mid + 1; else hi = mid; }
    const int start = lo;
    hi = N_ATOMS;
    while (lo < hi) { int mid = (lo + hi) >> 1; if (batch[mid] < g + 1) lo = mid + 1; else hi = mid; }
    const int end = lo;

    const int c0 = threadIdx.x, c1 = threadIdx.x + 256;
    float s0 = 0.0f, s1 = 0.0f;
    for (int r = start; r < end; ++r) {
        s0 += A[(long)r * HDIM + c0];
        s1 += A[(long)r * HDIM + c1];
    }
    gemb[(long)g * HDIM + c0] = s0;
    gemb[(long)g * HDIM + c1] = s1;
}

// ---------------------------------------------------------------------------
// graph_pred = relu(gemb @ W_g1 + b_g1) @ W_g2 + b_g2   [B x 1]
// WMMA version: 16 graphs per block (M-dim), 8 waves x 64 cols of the hidden
// layer each. W_g1 is read once per block (256 MB total L2 traffic instead of
// 4 GB for one-graph-per-block). Epilogue: relu * W_g2, 16-lane shuffle
// reduction, then cross-wave LDS reduction. Fixed order => deterministic.
// ---------------------------------------------------------------------------
__global__ __launch_bounds__(256) void graph_kernel(
        const float* __restrict__ gemb, const float* __restrict__ Wg1,
        const float* __restrict__ bg1, const float* __restrict__ Wg2,
        const float* __restrict__ bg2, float* __restrict__ out_graph) {
    __shared__ float red[8][16];
    const int tid  = threadIdx.x;
    const int wave = tid >> 5;
    const int lane = tid & 31;
    const int m    = lane & 15;
    const int koff = (lane >> 4) << 1;
    const int g0   = blockIdx.x * 16;     // B_GRAPHS/16 = 256 blocks
    const int nbase = wave * 64;          // 4 N-tiles of 16 per wave

    const float* Arow = gemb + (long)(g0 + m) * HDIM;

    v8f acc[4];
#pragma unroll
    for (int t = 0; t < 4; ++t) acc[t] = v8zero();

    for (int ks = 0; ks < 128; ++ks) {
        const int kl = ks * 4 + koff;
        v2f a = *reinterpret_cast<const v2f*>(Arow + kl);
#pragma unroll
        for (int nt = 0; nt < 4; ++nt) {
            const int col = nbase + nt * 16 + m;
            v2f b;
            b[0] = Wg1[(long)kl * HDIM + col];
            b[1] = Wg1[(long)(kl + 1) * HDIM + col];
            acc[nt] = wmma_f32_4(a, b, acc[nt]);
        }
    }

    // acc[nt][i]: graph = g0 + i + rbump, col = nbase + nt*16 + m
    const int rbump = (lane >> 4) << 3;
    float vsum[8];
#pragma unroll
    for (int i = 0; i < 8; ++i) vsum[i] = 0.0f;
#pragma unroll
    for (int nt = 0; nt < 4; ++nt) {
        const int col = nbase + nt * 16 + m;
        const float b1 = bg1[col];
        const float w2 = Wg2[col];
#pragma unroll
        for (int i = 0; i < 8; ++i) {
            float h = acc[nt][i] + b1;
            h = h > 0.0f ? h : 0.0f;
            vsum[i] += h * w2;
        }
    }
    // reduce over the 16 lanes of each half-wave (cols); xor masks 1..8 stay
    // inside each 16-lane group.
#pragma unroll
    for (int i = 0; i < 8; ++i) {
#pragma unroll
        for (int off = 1; off < 16; off <<= 1)
            vsum[i] += __shfl_xor(vsum[i], off, 32);
    }
    if (m == 0) {   // lane 0 holds graphs 0..7, lane 16 holds graphs 8..15
#pragma unroll
        for (int i = 0; i < 8; ++i)
            red[wave][i + rbump] = vsum[i];
    }
    __syncthreads();
    if (tid < 16) {
        float s = 0.0f;
#pragma unroll
        for (int w = 0; w < 8; ++w) s += red[w][tid];
        out_graph[g0 + tid] = s + bg2[0];
    }
}

// ---------------------------------------------------------------------------
extern "C" void kernel_launch(void* const* d_in, const int* in_sizes, int n_in,
                              void* d_out, int out_size, void* d_ws, size_t ws_size,
                              hipStream_t stream) {
    const float* A    = (const float*)d_in[0];
    const float* Wn   = (const float*)d_in[1];
    const float* bn   = (const float*)d_in[2];
    const float* We   = (const float*)d_in[3];
    const float* be   = (const float*)d_in[4];
    const float* Wg1  = (const float*)d_in[5];
    const float* bg1  = (const float*)d_in[6];
    const float* Wg2  = (const float*)d_in[7];
    const float* bg2  = (const float*)d_in[8];
    const int*   eidx = (const int*)d_in[9];
    const int*   batch= (const int*)d_in[11];

    float* out       = (float*)d_out;
    float* out_node  = out;                                        // N*AF
    float* out_edge  = out + (size_t)N_ATOMS * AF;                 // (E/2)*BF
    float* out_graph = out_edge + (size_t)(E_EDGES / 2) * BF;      // B
    float* gemb      = (float*)d_ws;                               // B*H floats (8 MB)

    node_kernel<<<(N_ATOMS + 127) / 128, 256, 0, stream>>>(A, Wn, bn, out_node);
    edge_kernel<<<(E_EDGES / 2) / 128, 256, 0, stream>>>(A, eidx, We, be, out_edge);
    segsum_kernel<<<B_GRAPHS, 256, 0, stream>>>(A, batch, gemb);
    graph_kernel<<<B_GRAPHS / 16, 256, 0, stream>>>(gemb, Wg1, bg1, Wg2, bg2, out_graph);
}